// BaseModule_57114475102494
// MI455X (gfx1250) — compile-verified
//
#include <hip/hip_runtime.h>

// Problem constants (match reference).
#define BB 2
#define VV 4
#define CC 64
#define HH 256
#define WW 256
static constexpr int   NPix    = HH * WW;            // 65536 pixels per image
static constexpr int   BV      = BB * VV;            // 8 images
static constexpr long  NPTOT   = (long)BV * NPix;    // 524288 source pixels
static constexpr float EPSF    = 1e-8f;
static constexpr float INV_RAD = 0.5f;               // 1/RADIUS, RADIUS=2

typedef __attribute__((ext_vector_type(2))) float v2f;
typedef __attribute__((ext_vector_type(8))) float v8f;

// ---------------------------------------------------------------- utilities
__device__ inline void mul4(const float* a, const float* b, float* c) {
    for (int i = 0; i < 4; ++i)
        for (int j = 0; j < 4; ++j) {
            float s = 0.f;
            for (int k = 0; k < 4; ++k) s += a[i * 4 + k] * b[k * 4 + j];
            c[i * 4 + j] = s;
        }
}

// General 4x4 inverse (adjugate / cofactor, MESA-style).
__device__ inline void inv4x4(const float* m, float* o) {
    float inv[16];
    inv[0]  =  m[5]*m[10]*m[15] - m[5]*m[11]*m[14] - m[9]*m[6]*m[15] + m[9]*m[7]*m[14] + m[13]*m[6]*m[11] - m[13]*m[7]*m[10];
    inv[4]  = -m[4]*m[10]*m[15] + m[4]*m[11]*m[14] + m[8]*m[6]*m[15] - m[8]*m[7]*m[14] - m[12]*m[6]*m[11] + m[12]*m[7]*m[10];
    inv[8]  =  m[4]*m[9]*m[15]  - m[4]*m[11]*m[13] - m[8]*m[5]*m[15] + m[8]*m[7]*m[13] + m[12]*m[5]*m[11] - m[12]*m[7]*m[9];
    inv[12] = -m[4]*m[9]*m[14]  + m[4]*m[10]*m[13] + m[8]*m[5]*m[14] - m[8]*m[6]*m[13] - m[12]*m[5]*m[10] + m[12]*m[6]*m[9];
    inv[1]  = -m[1]*m[10]*m[15] + m[1]*m[11]*m[14] + m[9]*m[2]*m[15] - m[9]*m[3]*m[14] - m[13]*m[2]*m[11] + m[13]*m[3]*m[10];
    inv[5]  =  m[0]*m[10]*m[15] - m[0]*m[11]*m[14] - m[8]*m[2]*m[15] + m[8]*m[3]*m[14] + m[12]*m[2]*m[11] - m[12]*m[3]*m[10];
    inv[9]  = -m[0]*m[9]*m[15]  + m[0]*m[11]*m[13] + m[8]*m[1]*m[15] - m[8]*m[3]*m[13] - m[12]*m[1]*m[11] + m[12]*m[3]*m[9];
    inv[13] =  m[0]*m[9]*m[14]  - m[0]*m[10]*m[13] - m[8]*m[1]*m[14] + m[8]*m[2]*m[13] + m[12]*m[1]*m[10] - m[12]*m[2]*m[9];
    inv[2]  =  m[1]*m[6]*m[15]  - m[1]*m[7]*m[14]  - m[5]*m[2]*m[15] + m[5]*m[3]*m[14] + m[13]*m[2]*m[7]  - m[13]*m[3]*m[6];
    inv[6]  = -m[0]*m[6]*m[15]  + m[0]*m[7]*m[14]  + m[4]*m[2]*m[15] - m[4]*m[3]*m[14] - m[12]*m[2]*m[7]  + m[12]*m[3]*m[6];
    inv[10] =  m[0]*m[5]*m[15]  - m[0]*m[7]*m[13]  - m[4]*m[1]*m[15] + m[4]*m[3]*m[13] + m[12]*m[1]*m[7]  - m[12]*m[3]*m[5];
    inv[14] = -m[0]*m[5]*m[14]  + m[0]*m[6]*m[13]  + m[4]*m[1]*m[14] - m[4]*m[2]*m[13] - m[12]*m[1]*m[6]  + m[12]*m[2]*m[5];
    inv[3]  = -m[1]*m[6]*m[11]  + m[1]*m[7]*m[10]  + m[5]*m[2]*m[11] - m[5]*m[3]*m[10] - m[9]*m[2]*m[7]   + m[9]*m[3]*m[6];
    inv[7]  =  m[0]*m[6]*m[11]  - m[0]*m[7]*m[10]  - m[4]*m[2]*m[11] + m[4]*m[3]*m[10] + m[8]*m[2]*m[7]   - m[8]*m[3]*m[6];
    inv[11] = -m[0]*m[5]*m[11]  + m[0]*m[7]*m[9]   + m[4]*m[1]*m[11] - m[4]*m[3]*m[9]  - m[8]*m[1]*m[7]   + m[8]*m[3]*m[5];
    inv[15] =  m[0]*m[5]*m[10]  - m[0]*m[6]*m[9]   - m[4]*m[1]*m[10] + m[4]*m[2]*m[9]  + m[8]*m[1]*m[6]   - m[8]*m[2]*m[5];
    float det = m[0]*inv[0] + m[1]*inv[4] + m[2]*inv[8] + m[3]*inv[12];
    float id = 1.0f / det;
    for (int i = 0; i < 16; ++i) o[i] = inv[i] * id;
}

// --------------------------------------------------- kernel 0: build M per (b,v)
// M = K @ dst_RT @ src_RTinv @ inv(K)
__global__ void setup_M_kernel(const float* __restrict__ K,
                               const float* __restrict__ src_RTinvs,
                               const float* __restrict__ dst_RTs,
                               float* __restrict__ Mout) {
    int bv = threadIdx.x;
    if (bv >= BV) return;
    int b = bv / VV;
    float Kinv[16], t1[16], t2[16], M[16];
    inv4x4(K + b * 16, Kinv);
    mul4(src_RTinvs + bv * 16, Kinv, t1);   // src_RTinv @ Kinv
    mul4(dst_RTs + b * 16, t1, t2);         // dst_RT @ ...
    mul4(K + b * 16, t2, M);                // K @ ...
    for (int i = 0; i < 16; ++i) Mout[bv * 16 + i] = M[i];
}

// --------------------------------------------------- kernel 1: zero fill
__global__ void fillz_kernel(float* __restrict__ p, long n) {
    long i = (long)blockIdx.x * blockDim.x + threadIdx.x;
    long stride = (long)gridDim.x * blockDim.x;
    for (; i < n; i += stride) p[i] = 0.0f;
}

// --------------------------------------------------- kernel 2: WMMA projection
// Each wave32 projects 32 pixels via two V_WMMA_F32_16X16X4_F32 ops:
//   D(16x16) = A(16x4 = M padded) x B(4x16 = proj columns for 16 pixels)
// ISA f32 layouts: A: v[0]=(M=lane%16,K=0|2), v[1]=(K=1|3);  B mirrored (K rows,
// N=lane%16 cols); D: VGPR r = row r on lanes 0-15 (rows 8-15 on lanes 16-31).
__global__ void proj_kernel(const float* __restrict__ depths,
                            const float* __restrict__ Mmat,
                            float* __restrict__ xo,
                            float* __restrict__ yo,
                            float* __restrict__ zo) {
    const int lane   = threadIdx.x & 31;
    const long waveId = ((long)blockIdx.x * blockDim.x + threadIdx.x) >> 5;
    const long basePix = waveId * 32;                 // NPix multiple of 32
    const int  n  = lane & 15;                        // matrix column / row id
    const int  k0 = (lane >> 4) * 2;                  // K pair this half-wave owns
    const int  bv = (int)(basePix >> 16);             // NPix == 65536

    // A operand: rows 0-3 = M, rows 4-15 = 0
    v2f a;
    a.x = (n < 4) ? Mmat[bv * 16 + n * 4 + k0]     : 0.0f;
    a.y = (n < 4) ? Mmat[bv * 16 + n * 4 + k0 + 1] : 0.0f;

    #pragma unroll
    for (int j = 0; j < 2; ++j) {
        long p   = basePix + j * 16 + n;              // pixel this column feeds
        int  pin = (int)(p & (NPix - 1));
        float gx = (float)(pin & (WW - 1));
        float gy = (float)(pin >> 8);                 // WW == 256
        float dep = depths[p];
        // B columns: proj = (gx*d, gy*d, d, 1)
        v2f bop;
        bop.x = (k0 == 0) ? gx * dep : dep;
        bop.y = (k0 == 0) ? gy * dep : 1.0f;
        v8f c = {};
        v8f d = __builtin_amdgcn_wmma_f32_16x16x4_f32(
            /*neg_a=*/false, a, /*neg_b=*/false, bop,
            /*c_mod=*/(short)0, c, /*reuse_a=*/false, /*reuse_b=*/false);
        if (lane < 16) {                              // rows 0..2 live here
            float z    = d[2];
            float invz = 1.0f / (z + EPSF);
            long  q    = basePix + j * 16 + lane;
            xo[q] = d[0] * invz;
            yo[q] = d[1] * invz;
            zo[q] = z;
        }
    }
}

// --------------------------------------------------- kernel 3: 5x5 splat scatter
// 2 threads per source pixel (32 channels each). Features accumulate straight
// into d_out channels 0..63; (wsum, w*z) into ws. Accumulator fits in 192MB L2.
#define CCHUNK 32
__global__ void splat_kernel(const float* __restrict__ feats,
                             const float* __restrict__ xs,
                             const float* __restrict__ ys,
                             const float* __restrict__ zs,
                             float* __restrict__ out,
                             float* __restrict__ wz) {
    long tid = (long)blockIdx.x * blockDim.x + threadIdx.x;
    if (tid >= NPTOT * 2) return;
    const int  chunk = (int)(tid / NPTOT);            // adjacent tids -> adjacent pixels
    const long p     = tid % NPTOT;
    const int  bv    = (int)(p >> 16);
    const int  pin   = (int)(p & (NPix - 1));

    const float x = xs[p], y = ys[p], z = zs[p];
    const float invz = 1.0f / (z + EPSF);
    const float px0 = rintf(x), py0 = rintf(y);
    const bool  zok = (z > EPSF);

    float f[CCHUNK];
    const float* fp = feats + ((long)bv * CC + chunk * CCHUNK) * NPix + pin;
    #pragma unroll
    for (int c = 0; c < CCHUNK; ++c) f[c] = fp[(long)c * NPix];

    float* obase = out + ((long)bv * (CC + 1) + chunk * CCHUNK) * NPix;

    #pragma unroll 1
    for (int dy = -2; dy <= 2; ++dy) {
        #pragma unroll 1
        for (int dx = -2; dx <= 2; ++dx) {
            const float pxf = px0 + (float)dx;
            const float pyf = py0 + (float)dy;
            const float ex = pxf - x, ey = pyf - y;
            const float dist = sqrtf(ex * ex + ey * ey + EPSF);
            float w = fmaxf(0.0f, 1.0f - dist * INV_RAD);
            const bool valid = (pxf >= 0.0f) && (pxf < (float)WW) &&
                               (pyf >= 0.0f) && (pyf < (float)HH) && zok;
            w = valid ? w * invz : 0.0f;
            if (w > 0.0f) {
                int ix = (int)pxf; ix = ix < 0 ? 0 : (ix > WW - 1 ? WW - 1 : ix);
                int iy = (int)pyf; iy = iy < 0 ? 0 : (iy > HH - 1 ? HH - 1 : iy);
                const long dpix = (long)iy * WW + ix;
                float* op = obase + dpix;
                #pragma unroll
                for (int c = 0; c < CCHUNK; ++c)
                    atomicAdd(op + (long)c * NPix, w * f[c]);
                if (chunk == 0) {
                    float* wp = wz + ((long)bv * NPix + dpix) * 2;
                    atomicAdd(wp,     w);
                    atomicAdd(wp + 1, w * z);
                }
            }
        }
    }
}

// --------------------------------------------------- kernel 4: normalize
__global__ void norm_kernel(float* __restrict__ out,
                            const float* __restrict__ wz) {
    const long total = (long)BV * (CC + 1) * NPix;
    long i = (long)blockIdx.x * blockDim.x + threadIdx.x;
    if (i >= total) return;
    const long p  = i % NPix;
    const long t  = i / NPix;
    const int  ch = (int)(t % (CC + 1));
    const long bv = t / (CC + 1);
    const float* wp = wz + (bv * NPix + p) * 2;
    const float inv = 1.0f / (wp[0] + EPSF);
    out[i] = (ch == CC) ? wp[1] * inv : out[i] * inv;
}

// --------------------------------------------------- host side
extern "C" void kernel_launch(void* const* d_in, const int* in_sizes, int n_in,
                              void* d_out, int out_size, void* d_ws, size_t ws_size,
                              hipStream_t stream) {
    const float* feats      = (const float*)d_in[0];
    const float* depths     = (const float*)d_in[1];
    const float* K          = (const float*)d_in[2];
    const float* src_RTinvs = (const float*)d_in[4];
    const float* dst_RTs    = (const float*)d_in[5];
    float* out = (float*)d_out;

    // workspace layout (floats): [M: 128][x: NP][y: NP][z: NP][wz: 2*NP]
    float* ws = (float*)d_ws;
    float* Mm = ws;
    float* xw = Mm + 128;
    float* yw = xw + NPTOT;
    float* zw = yw + NPTOT;
    float* wz = zw + NPTOT;

    const long outN = (long)BV * (CC + 1) * NPix;     // 34,078,720

    setup_M_kernel<<<1, BV, 0, stream>>>(K, src_RTinvs, dst_RTs, Mm);
    fillz_kernel<<<4096, 256, 0, stream>>>(out, outN);
    fillz_kernel<<<1024, 256, 0, stream>>>(wz, NPTOT * 2);

    // projection: one wave per 32 pixels -> NPTOT/32 waves
    const int projThreads = (int)(NPTOT);             // 32 threads per 32 pixels
    proj_kernel<<<projThreads / 256, 256, 0, stream>>>(depths, Mm, xw, yw, zw);

    const long splatThreads = NPTOT * 2;
    splat_kernel<<<(int)((splatThreads + 255) / 256), 256, 0, stream>>>(
        feats, xw, yw, zw, out, wz);

    norm_kernel<<<(int)((outN + 255) / 256), 256, 0, stream>>>(out, wz);
}